// Bleu_23390391894341
// MI455X (gfx1250) — compile-verified
//
#include <hip/hip_runtime.h>
#include <hip/hip_bf16.h>
#include <cstdint>
#include <cstddef>

// Problem constants (from reference setup): N=4, G=5, LEN_MAX=20, K=(G+1)*20+1=121
#define NGRAM 4
#define GREF  5
#define KCNT  121
#define NK    (NGRAM * KCNT)      // 484 floats per (b) of tc_res
#define GTSF  (GREF * NK)         // 2420 floats per (b) of tc_gts
#define LDSF  (GTSF + NK)         // 2904 floats staged per wave
#define WAVE  32
#define WPB   4                   // waves (batch rows) per block -> 46464 B LDS

#if defined(__gfx1250__) && __has_builtin(__builtin_amdgcn_global_load_async_to_lds_b128) && __has_builtin(__builtin_amdgcn_s_wait_asynccnt)
#define USE_ASYNC 1
#else
#define USE_ASYNC 0
#endif

#if USE_ASYNC
typedef int v4i_t __attribute__((ext_vector_type(4)));
typedef __attribute__((address_space(1))) v4i_t* as1_v4p;   // global int4*
typedef __attribute__((address_space(3))) v4i_t* as3_v4p;   // LDS int4*
__device__ __forceinline__ void async_copy_b128(const void* g, unsigned lds_byte_off) {
  // gfx1250 GLOBAL_LOAD_ASYNC_TO_LDS_B128: 16 bytes per lane, tracked by ASYNCcnt.
  // Pointers built via inttoptr: AS1 value = 64-bit VA, AS3 value = LDS byte offset.
  __builtin_amdgcn_global_load_async_to_lds_b128(
      (as1_v4p)(uintptr_t)g, (as3_v4p)(uintptr_t)lds_byte_off, 0, 0);
}
#endif

__device__ __forceinline__ float wave_sum(float v) {
#pragma unroll
  for (int s = WAVE / 2; s > 0; s >>= 1) v += __shfl_xor(v, s, WAVE);
  return v;
}

__global__ __launch_bounds__(WPB * WAVE) void bleu_kernel(
    const float* __restrict__ tc_res, const float* __restrict__ tc_gts,
    const float* __restrict__ l_res, const float* __restrict__ l_gts,
    float* __restrict__ out_score, float* __restrict__ out_list,
    float* __restrict__ out_judge, float* __restrict__ out_dif, int B) {
  const int lane = (int)(threadIdx.x & (WAVE - 1));
  const int wid  = (int)(threadIdx.x / WAVE);
  const int b    = (int)blockIdx.x * WPB + wid;
  if (b >= B) return;

#if USE_ASYNC
  __shared__ float lds[WPB][LDSF];
  {
    const float* gsrc = tc_gts + (size_t)b * GTSF;   // 16B aligned: 2420*4 = 16*605
    const float* rsrc = tc_res + (size_t)b * NK;     // 16B aligned: 484*4  = 16*121
    const unsigned base_off = (unsigned)(uintptr_t)(&lds[wid][0]);  // low 32b = LDS offset
    for (int t = lane; t < GTSF / 4; t += WAVE)
      async_copy_b128(gsrc + 4 * t, base_off + 16u * (unsigned)t);
    const unsigned roff = base_off + (unsigned)(GTSF * 4);
    for (int t = lane; t < NK / 4; t += WAVE)
      async_copy_b128(rsrc + 4 * t, roff + 16u * (unsigned)t);
    __builtin_amdgcn_s_wait_asynccnt(0);   // wave-private staging: no barrier needed
  }
  const float* gts = &lds[wid][0];
  const float* res = &lds[wid][GTSF];
#else
  const float* gts = tc_gts + (size_t)b * GTSF;
  const float* res = tc_res + (size_t)b * NK;
#endif

  // ---- clipped n-gram sums + g_judge -------------------------------------
  float clipped[NGRAM];
  float* judge = out_judge + (size_t)b * NK;
#pragma unroll
  for (int n = 0; n < NGRAM; ++n) {
    float partial = 0.0f;
    for (int k = lane; k < KCNT; k += WAVE) {
      const int off = n * KCNT + k;
      float gv = res[off];
      float m = gts[off];
#pragma unroll
      for (int g = 1; g < GREF; ++g) m = fmaxf(m, gts[g * NK + off]);
      const float mp = m + 1.0e-9f;      // same fp32 rounding as JAX (== m for m>=1)
      partial += fminf(gv, mp);
      // grad(min) != 0 iff gv < mp, or tie (0.5) => gv <= mp
      judge[off] = (gv <= mp) ? 1.0f : 0.0f;
    }
    clipped[n] = wave_sum(partial);      // broadcast to all lanes
  }

  // ---- wave-uniform length/brevity epilogue ------------------------------
  const float l_i = l_res[b];
  float best = 3.4e38f, l_r = 0.0f;
#pragma unroll
  for (int g = 0; g < GREF; ++g) {
    float lg = l_gts[(size_t)b * GREF + g];
    lg = (lg == 0.0f) ? 200.0f : lg;     // LEN_MAX*10
    const float d = fabsf(l_i - lg);
    if (d < best) { best = d; l_r = lg; }  // strict '<' => first argmin, as jnp.argmin
  }
  const float lie   = l_i + 1.0e-9f;
  const float ratio = (l_r + 1.0e-9f) / lie;
  const float mbr   = fmaxf(1.0f, ratio);   // fp32(1.0+1e-9) == 1.0f
  const float lf    = expf(1.0f - mbr);
  const float und   = 1.0f / lie;

  float ch0[NGRAM], ch1[NGRAM], ch2[NGRAM];
#pragma unroll
  for (int n = 0; n < NGRAM; ++n) {
    const float gf = clipped[n] * und;
    ch0[n] = gf;
    ch1[n] = gf + und;
    ch2[n] = fmaxf(gf - und, 0.0f);
  }
  const float base  = ch0[0] * ch0[1] * ch0[2] * ch0[3];
  const float qbase = sqrtf(sqrtf(base));   // base^(1/4), base >= 0

  // d/dl_i of exp(1 - max(1, ratio)) — JAX max splits ties 50/50 (ratio==1 is common)
  const float w   = (ratio > 1.0f) ? 1.0f : ((ratio == 1.0f) ? 0.5f : 0.0f);
  const float dif = qbase * lf * w * ((l_r + 1.0e-9f) / (lie * lie));
  if (lane == 0) {
    out_score[b] = lf * qbase;   // c_full[:,0]
    out_dif[b]   = dif;
  }

  const float c80 = lf * sqrtf(sqrtf(ch2[0] * ch2[1] * ch2[2] * ch2[3]));
  for (int t = lane; t < 81; t += WAVE) {
    const int i = t / 27, j = (t / 9) % 3, k = (t / 3) % 3, l = t % 3;
    const float p0 = (i == 0) ? ch0[0] : (i == 1) ? ch1[0] : ch2[0];
    const float p1 = (j == 0) ? ch0[1] : (j == 1) ? ch1[1] : ch2[1];
    const float p2 = (k == 0) ? ch0[2] : (k == 1) ? ch1[2] : ch2[2];
    const float p3 = (l == 0) ? ch0[3] : (l == 1) ? ch1[3] : ch2[3];
    out_list[(size_t)b * 81 + t] = lf * sqrtf(sqrtf(p0 * p1 * p2 * p3)) - c80;
  }
}

extern "C" void kernel_launch(void* const* d_in, const int* in_sizes, int n_in,
                              void* d_out, int out_size, void* d_ws, size_t ws_size,
                              hipStream_t stream) {
  (void)n_in; (void)out_size; (void)d_ws; (void)ws_size;
  const float* tc_res = (const float*)d_in[0];
  const float* tc_gts = (const float*)d_in[1];
  const float* l_res  = (const float*)d_in[2];
  const float* l_gts  = (const float*)d_in[3];
  const int B = in_sizes[2];                       // l_res has B*1 elements

  float* out       = (float*)d_out;
  float* out_score = out;                          // [B]
  float* out_list  = out + (size_t)B;              // [B,81]
  float* out_judge = out + (size_t)B * 82;         // [B,484] (bool -> 0/1 float)
  float* out_dif   = out + (size_t)B * 566;        // [B]

  const int blocks = (B + WPB - 1) / WPB;
  bleu_kernel<<<dim3(blocks), dim3(WPB * WAVE), 0, stream>>>(
      tc_res, tc_gts, l_res, l_gts, out_score, out_list, out_judge, out_dif, B);
}